// GCN_Entity_11888469475658
// MI455X (gfx1250) — compile-verified
//
#include <hip/hip_runtime.h>
#include <hip/hip_bf16.h>

// ---------------------------------------------------------------------------
// GCN layer for MI455X (gfx1250):
//   y = emb[nodes] @ W^T          (fp32 WMMA 16x16x4, double-buffered)
//   out = relu(segment_sum over edges+selfloops of y[src]*dinv[src]*dinv[dst] + b)
// ---------------------------------------------------------------------------

typedef float v2f __attribute__((ext_vector_type(2)));
typedef float v8f __attribute__((ext_vector_type(8)));

static constexpr int GDIM = 256;   // feature dim

// 32-bit byte-offset load from a uniform base -> saddr+voffset global_load_b64
__device__ __forceinline__ v2f ld2(const float* __restrict__ base, unsigned byteoff) {
  return *(const v2f*)((const char*)base + byteoff);
}

// ---------------- GEMM: y = emb[nodes] @ W^T via V_WMMA_F32_16X16X4_F32 ----
// One wave computes a 16(M) x 64(N) strip: 4 accumulator tiles.
// Pipeline: groups of 16 K (4 wmma-steps), ping-pong double buffered so the
// 20 loads of group i+1 are in flight during the 16 WMMAs of group i.
// Layouts (32-bit operands, wave32):
//   A 16x4 : lane l -> M = l%16, VGPR v holds K = 2*(l/16)+v
//   B 4x16 : lane l -> N = l%16, VGPR v holds K = 2*(l/16)+v   (B[k][j]=W[j][k])
//   C 16x16: lane l -> N = l%16, VGPR v holds M = v + 8*(l/16)
__global__ __launch_bounds__(128) void gcn_gemm_wmma(
    const int* __restrict__ nodes,
    const float* __restrict__ emb,   // [n, 256]
    const float* __restrict__ W,     // [256, 256] row-major
    float* __restrict__ y,           // [n, 256]
    int n)
{
  const int lane = threadIdx.x & 31;
  const int wave = threadIdx.x >> 5;
  const int half = lane >> 4;        // 0 or 1
  const int l16  = lane & 15;
  const int mtile = blockIdx.x * 4 + wave;   // 16-row tile per wave
  if (mtile * 16 >= n) return;               // uniform per wave; EXEC stays all-1
  const int strip = blockIdx.y;              // 64-column strip (0..3)

  // Per-lane 32-bit byte offsets of the k=0 fragment (+8*half bytes = 2*half floats)
  int rowA  = mtile * 16 + l16;
  int rowAc = rowA < n ? rowA : n - 1;
  const unsigned aoff = ((unsigned)nodes[rowAc] << 10) + (half << 3);
  const int j0 = strip * 64 + l16;
  unsigned boff[4];
  #pragma unroll
  for (int t = 0; t < 4; ++t) boff[t] = ((unsigned)(j0 + 16 * t) << 10) + (half << 3);

  v8f acc[4] = {{}, {}, {}, {}};

  v2f A0[4], A1[4];          // A frags, 4 k-steps per group
  v2f B0[4][4], B1[4][4];    // B frags [tile][k-step]

  auto loadgrp = [&](int k, v2f (&A)[4], v2f (&B)[4][4]) {
    #pragma unroll
    for (int u = 0; u < 4; ++u) {
      const unsigned kb = (unsigned)(k + 4 * u) << 2;   // byte offset of this k-step
      A[u] = ld2(emb, aoff + kb);
      #pragma unroll
      for (int t = 0; t < 4; ++t) B[t][u] = ld2(W, boff[t] + kb);
    }
  };
  auto compute = [&](v2f (&A)[4], v2f (&B)[4][4]) {
    #pragma unroll
    for (int u = 0; u < 4; ++u) {
      #pragma unroll
      for (int t = 0; t < 4; ++t)
        acc[t] = __builtin_amdgcn_wmma_f32_16x16x4_f32(
            false, A[u], false, B[t][u], (short)0, acc[t], false, false);
    }
  };

  loadgrp(0, A0, B0);
  #pragma unroll 1
  for (int k = 0; k < GDIM - 32; k += 32) {
    loadgrp(k + 16, A1, B1);
    compute(A0, B0);
    loadgrp(k + 32, A0, B0);
    compute(A1, B1);
  }
  loadgrp(GDIM - 16, A1, B1);
  compute(A0, B0);   // k = GDIM-32
  compute(A1, B1);   // k = GDIM-16

  // Store D tiles: row = mtile*16 + v + 8*half, col = strip*64 + t*16 + l16
  const int baseRow = mtile * 16 + 8 * half;
  #pragma unroll
  for (int v = 0; v < 8; ++v) {
    const int row = baseRow + v;
    if (row < n) {
      float* yr = y + (long)row * GDIM + strip * 64 + l16;
      yr[ 0] = acc[0][v];
      yr[16] = acc[1][v];
      yr[32] = acc[2][v];
      yr[48] = acc[3][v];
    }
  }
}

// ---------------- degree / normalization ----------------------------------
__global__ void zero_f32(float* __restrict__ p, int n) {
  int i = blockIdx.x * blockDim.x + threadIdx.x;
  if (i < n) p[i] = 0.0f;
}

__global__ void deg_count(const int* __restrict__ dst, float* __restrict__ deg, int E) {
  int e = blockIdx.x * blockDim.x + threadIdx.x;
  if (e < E) unsafeAtomicAdd(&deg[dst[e]], 1.0f);
}

__global__ void make_dinv(float* __restrict__ deg, int n) {
  int i = blockIdx.x * blockDim.x + threadIdx.x;
  if (i < n) deg[i] = rsqrtf(deg[i] + 1.0f);   // +1 = self-loop; deg>=1 always
}

// ---------------- self-loop init: out = y * dinv^2 -------------------------
__global__ void self_loop_init(const float* __restrict__ y,
                               const float* __restrict__ dinv,
                               float* __restrict__ out, int total) {
  int idx = blockIdx.x * blockDim.x + threadIdx.x;
  if (idx < total) {
    float d = dinv[idx >> 8];
    out[idx] = y[idx] * d * d;
  }
}

// ---------------- edge scatter: out[dst] += y[src] * norm ------------------
// thread t -> (edge e = t/64, chunk c = t%64 of 4 floats); float4 gather then
// 4 native f32 atomic adds (L2-resident: y + out both fit in 192MB L2).
__global__ void edge_scatter(const int* __restrict__ src,
                             const int* __restrict__ dstv,
                             const float* __restrict__ y,
                             const float* __restrict__ dinv,
                             float* __restrict__ out, int E) {
  int t = blockIdx.x * blockDim.x + threadIdx.x;
  int e = t >> 6;
  int c = t & 63;
  if (e < E) {
    int s = src[e], d = dstv[e];
    float norm = dinv[s] * dinv[d];
    const float4 v = ((const float4*)(y + (long)s * GDIM))[c];
    float* o = out + (long)d * GDIM + c * 4;
    unsafeAtomicAdd(o + 0, v.x * norm);
    unsafeAtomicAdd(o + 1, v.y * norm);
    unsafeAtomicAdd(o + 2, v.z * norm);
    unsafeAtomicAdd(o + 3, v.w * norm);
  }
}

// ---------------- bias + relu ----------------------------------------------
__global__ void bias_relu(float* __restrict__ out, const float* __restrict__ b, int total) {
  int idx = blockIdx.x * blockDim.x + threadIdx.x;
  if (idx < total) {
    float v = out[idx] + b[idx & (GDIM - 1)];
    out[idx] = v > 0.0f ? v : 0.0f;
  }
}

// ---------------------------------------------------------------------------
extern "C" void kernel_launch(void* const* d_in, const int* in_sizes, int n_in,
                              void* d_out, int out_size, void* d_ws, size_t ws_size,
                              hipStream_t stream) {
  const int*   nodes = (const int*)d_in[0];
  const int*   edges = (const int*)d_in[1];     // [2, E] flat: src then dst
  const float* emb   = (const float*)d_in[2];   // [n, 256]
  const float* W     = (const float*)d_in[3];   // [256, 256]
  const float* b     = (const float*)d_in[4];   // [256]
  float*       out   = (float*)d_out;           // [n, 256]

  const int n = in_sizes[0];
  const int E = in_sizes[1] / 2;
  const int total = n * GDIM;

  // workspace: y [n*256 f32] then deg/dinv [n f32]
  float* y   = (float*)d_ws;
  float* deg = y + (size_t)n * GDIM;

  zero_f32<<<(n + 255) / 256, 256, 0, stream>>>(deg, n);
  deg_count<<<(E + 255) / 256, 256, 0, stream>>>(edges + E, deg, E);
  make_dinv<<<(n + 255) / 256, 256, 0, stream>>>(deg, n);

  const int mtiles = (n + 15) / 16;
  dim3 g((mtiles + 3) / 4, GDIM / 64);
  gcn_gemm_wmma<<<g, 128, 0, stream>>>(nodes, emb, W, y, n);

  self_loop_init<<<(total + 255) / 256, 256, 0, stream>>>(y, deg, out, total);

  long sthreads = (long)E * 64;
  edge_scatter<<<(unsigned)((sthreads + 255) / 256), 256, 0, stream>>>(
      edges, edges + E, y, deg, out, E);

  bias_relu<<<(total + 255) / 256, 256, 0, stream>>>(out, b, total);
}